// BoxRepelLoss_19722489823319
// MI455X (gfx1250) — compile-verified
//
#include <hip/hip_runtime.h>
#include <hip/hip_bf16.h>
#include <math.h>

typedef __attribute__((ext_vector_type(2))) float v2f;
typedef __attribute__((ext_vector_type(8))) float v8f;

#define MBOX 768
#define REPEL_MARGIN 0.08f
#define MIN_SIZE 0.02f
#define IOU_MARGIN 0.1f
#define EPSV 1e-12f

// ---------------------------------------------------------------------------
// Kernel 1: pairwise center Gram matrix via V_WMMA_F32_16X16X4_F32.
// d2_ij = |ci|^2 + |cj|^2 - 2 * (ci . cj); repel = max(margin - sqrt(d2), 0).
// One wave (32 lanes) computes one 16x16 tile of the 768x768 matrix.
// A layout (16x4 f32): lanes 0-15 -> M=0..15, K={0,1}; lanes 16-31 -> K={2,3}
// (we zero K=2,3 since the true K is 2). B layout symmetric.
// D layout (16x16 f32, 8 VGPRs): lane L, VGPR v -> M = v + 8*(L>=16), N = L%16.
// ---------------------------------------------------------------------------
__global__ __launch_bounds__(256) void repel_wmma_kernel(
    const float* __restrict__ pred, float* __restrict__ partial) {
  const int wave = threadIdx.x >> 5;
  const int lane = threadIdx.x & 31;
  const int tile = blockIdx.x * 8 + wave;          // 0 .. 2303
  const int ti = tile / 48, tj = tile % 48;
  const int i0 = ti * 16, j0 = tj * 16;
  const int lane16 = lane & 15;
  const bool hi = lane >= 16;

  const float cxi = pred[(i0 + lane16) * 6 + 0];
  const float cyi = pred[(i0 + lane16) * 6 + 1];
  const float cxj = pred[(j0 + lane16) * 6 + 0];
  const float cyj = pred[(j0 + lane16) * 6 + 1];

  v2f A, B;
  A.x = hi ? 0.0f : cxi;  A.y = hi ? 0.0f : cyi;
  B.x = hi ? 0.0f : cxj;  B.y = hi ? 0.0f : cyj;
  v8f C = {};
  // 8 args: (neg_a, A, neg_b, B, c_mod, C, reuse_a, reuse_b)
  C = __builtin_amdgcn_wmma_f32_16x16x4_f32(
      false, A, false, B, (short)0, C, false, false);

  const float nj = cxj * cxj + cyj * cyj;          // column norm, j = j0+lane16
  const int rbase = i0 + (hi ? 8 : 0);
  float lsum = 0.0f;
#pragma unroll
  for (int v = 0; v < 8; ++v) {
    const int gi = rbase + v;
    const float rx = pred[gi * 6 + 0];
    const float ry = pred[gi * 6 + 1];
    const float ni = rx * rx + ry * ry;
    float d2 = fmaxf(ni + nj - 2.0f * C[v], 0.0f);
    float rep = fmaxf(REPEL_MARGIN - sqrtf(d2), 0.0f);
    lsum += (gi != (j0 + lane16)) ? rep : 0.0f;    // mask diagonal
  }
  // wave32 reduction
  for (int off = 16; off > 0; off >>= 1)
    lsum += __shfl_down(lsum, off, 32);
  if (lane == 0) partial[tile] = lsum;
}

// ---------------------------------------------------------------------------
// Kernel 2: rotated-box IoU penalty, one thread per (i,j) pair.
// ---------------------------------------------------------------------------
struct OBB { float cx, cy, w, h, c, s; };

__device__ __forceinline__ OBB load_box(const float* __restrict__ p, int k) {
  OBB b;
  b.cx = p[k * 6 + 0]; b.cy = p[k * 6 + 1];
  b.w  = p[k * 6 + 2]; b.h  = p[k * 6 + 3];
  float c = p[k * 6 + 4], s = p[k * 6 + 5];
  float inv = rsqrtf(fmaxf(c * c + s * s, 1e-30f)); // ref: atan2 -> cos/sin
  b.c = c * inv; b.s = s * inv;
  return b;
}

__device__ __forceinline__ void obb_corners(const OBB& b, float* x, float* y) {
  const float hw = 0.5f * b.w, hh = 0.5f * b.h;
  const float dx[4] = {-hw, hw, hw, -hw};
  const float dy[4] = {-hh, -hh, hh, hh};
#pragma unroll
  for (int k = 0; k < 4; ++k) {
    x[k] = b.cx + b.c * dx[k] - b.s * dy[k];
    y[k] = b.cy + b.s * dx[k] + b.c * dy[k];
  }
}

__device__ __forceinline__ bool in_box(float px, float py, const OBB& b) {
  const float dx = px - b.cx, dy = py - b.cy;
  const float lx = b.c * dx + b.s * dy;
  const float ly = -b.s * dx + b.c * dy;
  return (fabsf(lx) <= 0.5f * b.w) && (fabsf(ly) <= 0.5f * b.h);
}

__device__ float pair_iou(const OBB& b1, const OBB& b2) {
  float x1[4], y1[4], x2[4], y2[4];
  obb_corners(b1, x1, y1);
  obb_corners(b2, x2, y2);

  float px[24], py[24];
  int n = 0;
  // segment-segment intersections (e1-major order, matches reference reshape)
#pragma unroll
  for (int e1 = 0; e1 < 4; ++e1) {
    const float rx = x1[(e1 + 1) & 3] - x1[e1];
    const float ry = y1[(e1 + 1) & 3] - y1[e1];
#pragma unroll
    for (int e2 = 0; e2 < 4; ++e2) {
      const float sx = x2[(e2 + 1) & 3] - x2[e2];
      const float sy = y2[(e2 + 1) & 3] - y2[e2];
      const float den = rx * sy - ry * sx;
      if (fabsf(den) > EPSV) {
        const float qpx = x2[e2] - x1[e1];
        const float qpy = y2[e2] - y1[e1];
        const float t = (qpx * sy - qpy * sx) / den;
        const float u = (qpx * ry - qpy * rx) / den;
        if (t >= 0.0f && t <= 1.0f && u >= 0.0f && u <= 1.0f) {
          px[n] = x1[e1] + t * rx;
          py[n] = y1[e1] + t * ry;
          ++n;
        }
      }
    }
  }
  // corners of box2 inside box1, then corners of box1 inside box2
#pragma unroll
  for (int k = 0; k < 4; ++k)
    if (in_box(x2[k], y2[k], b1)) { px[n] = x2[k]; py[n] = y2[k]; ++n; }
#pragma unroll
  for (int k = 0; k < 4; ++k)
    if (in_box(x1[k], y1[k], b2)) { px[n] = x1[k]; py[n] = y1[k]; ++n; }

  float sx = 0.0f, sy = 0.0f;
  for (int k = 0; k < n; ++k) { sx += px[k]; sy += py[k]; }
  const float invc = 1.0f / (float)max(n, 1);
  const float mx = sx * invc, my = sy * invc;

  // stable insertion sort by angle around the mean
  float ang[24];
  for (int k = 0; k < n; ++k) ang[k] = atan2f(py[k] - my, px[k] - mx);
  for (int k = 1; k < n; ++k) {
    const float a = ang[k], X = px[k], Y = py[k];
    int l = k - 1;
    while (l >= 0 && ang[l] > a) {
      ang[l + 1] = ang[l]; px[l + 1] = px[l]; py[l + 1] = py[l];
      --l;
    }
    ang[l + 1] = a; px[l + 1] = X; py[l + 1] = Y;
  }
  // shoelace around the mean
  float acc = 0.0f;
  for (int k = 0; k < n; ++k) {
    const int k2 = (k + 1 == n) ? 0 : k + 1;
    acc += (px[k] - mx) * (py[k2] - my) - (py[k] - my) * (px[k2] - mx);
  }
  const float inter = 0.5f * fabsf(acc);
  const float uni = b1.w * b1.h + b2.w * b2.h - inter;
  return (uni > EPSV) ? inter / fmaxf(uni, EPSV) : 0.0f;
}

__global__ __launch_bounds__(128) void iou_kernel(
    const float* __restrict__ pred, float* __restrict__ partial) {
  const int idx = blockIdx.x * 128 + threadIdx.x;   // 0 .. 768*768-1
  const int i = idx / MBOX;
  const int j = idx - i * MBOX;
  float contrib = 0.0f;
  if (i != j) {
    const OBB b1 = load_box(pred, i);
    const OBB b2 = load_box(pred, j);
    contrib = fmaxf(pair_iou(b1, b2) - IOU_MARGIN, 0.0f);
  }
  __shared__ float red[128];
  red[threadIdx.x] = contrib;
  __syncthreads();
  for (int s = 64; s > 0; s >>= 1) {
    if (threadIdx.x < s) red[threadIdx.x] += red[threadIdx.x + s];
    __syncthreads();
  }
  if (threadIdx.x == 0) partial[blockIdx.x] = red[0];
}

// ---------------------------------------------------------------------------
// Kernel 3: deterministic final reduction + size penalty.
// ---------------------------------------------------------------------------
__global__ __launch_bounds__(256) void finalize_kernel(
    const float* __restrict__ pred,
    const float* __restrict__ repel_partial, int n_repel,
    const float* __restrict__ iou_partial, int n_iou,
    float* __restrict__ out) {
  __shared__ float sm[256];
  __shared__ float total;
  const int tid = threadIdx.x;
  if (tid == 0) total = 0.0f;
  __syncthreads();

  // repel
  float s = 0.0f;
  for (int k = tid; k < n_repel; k += 256) s += repel_partial[k];
  sm[tid] = s; __syncthreads();
  for (int w = 128; w > 0; w >>= 1) {
    if (tid < w) sm[tid] += sm[tid + w];
    __syncthreads();
  }
  if (tid == 0) total += sm[0] / ((float)MBOX * (float)(MBOX - 1));
  __syncthreads();

  // size penalty
  s = 0.0f;
  for (int k = tid; k < MBOX; k += 256)
    s += fmaxf(MIN_SIZE - pred[k * 6 + 2], 0.0f) +
         fmaxf(MIN_SIZE - pred[k * 6 + 3], 0.0f);
  sm[tid] = s; __syncthreads();
  for (int w = 128; w > 0; w >>= 1) {
    if (tid < w) sm[tid] += sm[tid + w];
    __syncthreads();
  }
  if (tid == 0) total += sm[0] / (float)MBOX;
  __syncthreads();

  // iou
  s = 0.0f;
  for (int k = tid; k < n_iou; k += 256) s += iou_partial[k];
  sm[tid] = s; __syncthreads();
  for (int w = 128; w > 0; w >>= 1) {
    if (tid < w) sm[tid] += sm[tid + w];
    __syncthreads();
  }
  if (tid == 0) {
    total += sm[0] / ((float)MBOX * (float)MBOX);
    out[0] = total;
  }
}

extern "C" void kernel_launch(void* const* d_in, const int* in_sizes, int n_in,
                              void* d_out, int out_size, void* d_ws, size_t ws_size,
                              hipStream_t stream) {
  (void)in_sizes; (void)n_in; (void)out_size; (void)ws_size;
  const float* pred = (const float*)d_in[0];   // (769, 6) f32; we use rows 0..767
  float* out = (float*)d_out;

  float* repel_partial = (float*)d_ws;         // 2304 floats
  float* iou_partial = repel_partial + 2304;   // 4608 floats

  // 48x48 = 2304 tiles, 8 waves per 256-thread block -> 288 blocks
  repel_wmma_kernel<<<288, 256, 0, stream>>>(pred, repel_partial);
  // 768*768 pairs / 128 = 4608 blocks
  iou_kernel<<<4608, 128, 0, stream>>>(pred, iou_partial);
  finalize_kernel<<<1, 256, 0, stream>>>(pred, repel_partial, 2304,
                                         iou_partial, 4608, out);
}